// GroupedQueryAttention_22076131901743
// MI455X (gfx1250) — compile-verified
//
#include <hip/hip_runtime.h>
#include <hip/hip_bf16.h>
#include <math.h>

typedef __attribute__((ext_vector_type(2))) float v2f;
typedef __attribute__((ext_vector_type(4))) float v4f;
typedef __attribute__((ext_vector_type(8))) float v8f;
typedef __attribute__((ext_vector_type(4))) unsigned int u32x4;
typedef __attribute__((ext_vector_type(4))) int i32x4;
typedef __attribute__((ext_vector_type(8))) int i32x8;

#define B_    2
#define S_    2048
#define D_    2048
#define H_    16
#define KVH_  4
#define HD_   128
#define GROUPS_ (H_ / KVH_)

static __device__ inline v8f v8zero() {
  v8f v = {0.f, 0.f, 0.f, 0.f, 0.f, 0.f, 0.f, 0.f};
  return v;
}

// ---------------------------------------------------------------------------
// TDM: issue a 2D tensor-tile load (global -> LDS) via the Tensor Data Mover.
// D# built per CDNA5 ISA 8.3/8.4:
//   group0: count=1 | lds_addr | global_addr[56:0] | type=2
//   group1: data_size=4B, optional LDS padding, tensor dims, tile dims, stride
// pad_icode: 3 -> every 16 DWORDs, 6 -> every 128 DWORDs; pad_acode: 1 -> +2 DW
// ---------------------------------------------------------------------------
static __device__ inline void tdm_load_2d(unsigned lds_byte_addr,
                                          const void* gaddr,
                                          unsigned tile_d0, unsigned tile_d1,
                                          unsigned tensor_d0, unsigned tensor_d1,
                                          unsigned stride_d0,
                                          unsigned pad_icode, unsigned pad_acode)
{
  unsigned long long ga = (unsigned long long)(size_t)gaddr;
  u32x4 g0;
  g0[0] = 1u;                                              // count=1 (valid)
  g0[1] = lds_byte_addr;                                   // bits 63:32
  g0[2] = (unsigned)(ga & 0xFFFFFFFFu);                    // global_addr lo
  g0[3] = (unsigned)((ga >> 32) & 0x01FFFFFFu) | (2u << 30); // addr hi | type=2
  i32x8 g1;
  unsigned w0 = (2u << 16)                                  // data_size = 4B
              | (1u << 20)                                  // pad_enable
              | (pad_icode << 22) | (pad_acode << 25);
  g1[0] = (int)w0;
  g1[1] = (int)((tensor_d0 & 0xFFFFu) << 16);               // tensor_dim0[15:0]
  g1[2] = (int)((tensor_d0 >> 16) | ((tensor_d1 & 0xFFFFu) << 16));
  g1[3] = (int)((tensor_d1 >> 16) | ((tile_d0 & 0xFFFFu) << 16));
  g1[4] = (int)(tile_d1 & 0xFFFFu);                         // tile_dim1 (tile_dim2=0)
  g1[5] = (int)stride_d0;                                   // tensor_dim0_stride lo
  g1[6] = 0;
  g1[7] = 0;
  i32x4 z4 = {0, 0, 0, 0};
#if defined(__clang_major__) && (__clang_major__ >= 23)
  i32x8 z8 = {0, 0, 0, 0, 0, 0, 0, 0};
  __builtin_amdgcn_tensor_load_to_lds(g0, g1, z4, z4, z8, 0);
#else
  __builtin_amdgcn_tensor_load_to_lds(g0, g1, z4, z4, 0);
#endif
}

// ---------------------------------------------------------------------------
// Kernel 1: fused QKV projection + RoPE, fp32 WMMA, TDM double-buffered tiles.
// grid = (B*S/128, 24); block = 256 (8 wave32)
//   colBlock 0..15  -> Q head cb       (RoPE)
//   colBlock 16..19 -> K kv-head cb-16 (RoPE)
//   colBlock 20..23 -> V kv-head cb-20 (no RoPE)
// Wave strip = 16 rows x 128 cols (full head) so rotate-half partner
// (d ^ 64 <=> d-tile t ^ 4) is in the same lane's registers.
// ---------------------------------------------------------------------------
__global__ __launch_bounds__(256)
void qkv_rope_kernel(const float* __restrict__ hidden,
                     const float* __restrict__ cosT,
                     const float* __restrict__ sinT,
                     const float* __restrict__ Wq,
                     const float* __restrict__ Wk,
                     const float* __restrict__ Wv,
                     float* __restrict__ q_out,
                     float* __restrict__ k_out,
                     float* __restrict__ v_out)
{
  __shared__ float Alds[2][128][18];   // 128 rows x 16 K, TDM-padded 16->18
  __shared__ float Blds[2][16][130];   // 16 K x 128 N, TDM-padded 128->130

  const int tid  = threadIdx.x;
  const int wave = tid >> 5;
  const int lane = tid & 31;
  const int lo   = lane & 15;
  const int hi   = lane >> 4;

  const int rowbase = blockIdx.x * 128;
  const int cb      = blockIdx.y;

  const float* W;
  float* outp;
  int ldw, head, nheads, doRope;
  if (cb < 16)      { W = Wq + cb * 128;        ldw = H_ * HD_;   head = cb;      nheads = H_;   doRope = 1; outp = q_out; }
  else if (cb < 20) { W = Wk + (cb - 16) * 128; ldw = KVH_ * HD_; head = cb - 16; nheads = KVH_; doRope = 1; outp = k_out; }
  else              { W = Wv + (cb - 20) * 128; ldw = KVH_ * HD_; head = cb - 20; nheads = KVH_; doRope = 0; outp = v_out; }

  v8f acc[8];
  #pragma unroll
  for (int t = 0; t < 8; ++t) acc[t] = v8zero();

  const int nchunks = D_ / 16;   // 128 chunks of K=16

  // prologue: TDM-load chunk 0 into buffer 0 (wave 0 owns TENSORcnt)
  if (wave == 0) {
    tdm_load_2d((unsigned)(size_t)&Alds[0][0][0], hidden + (size_t)rowbase * D_,
                /*tile*/16, 128, /*tensor*/D_, B_ * S_, /*stride*/D_,
                /*pad: every 16 DW*/3, /*+2 DW*/1);
    tdm_load_2d((unsigned)(size_t)&Blds[0][0][0], W,
                /*tile*/128, 16, /*tensor*/(unsigned)ldw, D_, (unsigned)ldw,
                /*pad: every 128 DW*/6, /*+2 DW*/1);
  }

  for (int ch = 0; ch < nchunks; ++ch) {
    const int buf = ch & 1;
    if (wave == 0) {
      if (ch + 1 < nchunks) {   // prefetch next chunk into the other buffer
        const int kb = (ch + 1) * 16;
        tdm_load_2d((unsigned)(size_t)&Alds[buf ^ 1][0][0],
                    hidden + (size_t)rowbase * D_ + kb,
                    16, 128, D_, B_ * S_, D_, 3, 1);
        tdm_load_2d((unsigned)(size_t)&Blds[buf ^ 1][0][0],
                    W + (size_t)kb * ldw,
                    128, 16, (unsigned)ldw, D_, (unsigned)ldw, 6, 1);
        __builtin_amdgcn_s_wait_tensorcnt(2);   // current pair done, next in flight
      } else {
        __builtin_amdgcn_s_wait_tensorcnt(0);
      }
    }
    __syncthreads();   // current buffer ready for all waves

    #pragma unroll
    for (int kk = 0; kk < 4; ++kk) {
      const int k = kk * 4 + hi * 2;
      v2f a;                                    // A 16x4: lane = m + 16*(k>>1)
      a.x = Alds[buf][wave * 16 + lo][k];
      a.y = Alds[buf][wave * 16 + lo][k + 1];
      #pragma unroll
      for (int t = 0; t < 8; ++t) {
        v2f b;                                  // B 4x16 (mirrored A layout)
        b.x = Blds[buf][k][t * 16 + lo];
        b.y = Blds[buf][k + 1][t * 16 + lo];
        acc[t] = __builtin_amdgcn_wmma_f32_16x16x4_f32(
            false, a, false, b, (short)0, acc[t], false, false);
      }
    }
    __syncthreads();   // all reads of this buffer done before TDM overwrites it
  }

  // --- epilogue: RoPE in registers, write out ---
  #pragma unroll
  for (int r = 0; r < 8; ++r) {
    const int m   = r + 8 * hi;
    const int row = rowbase + wave * 16 + m;
    const int b   = row / S_;
    const int s   = row % S_;
    float* orow = outp + (((size_t)b * nheads + head) * S_ + s) * HD_;
    if (doRope) {
      #pragma unroll
      for (int t = 0; t < 8; ++t) {
        const int d  = t * 16 + lo;
        const float c  = cosT[(size_t)s * HD_ + d];
        const float sn = sinT[(size_t)s * HD_ + d];
        const float v0 = acc[t][r];
        const float v1 = acc[t ^ 4][r];         // rotate-half partner (d ^ 64)
        orow[d] = (t < 4) ? (v0 * c - v1 * sn) : (v0 * c + v1 * sn);
      }
    } else {
      #pragma unroll
      for (int t = 0; t < 8; ++t) orow[t * 16 + lo] = acc[t][r];
    }
  }
}

// ---------------------------------------------------------------------------
// Kernel 2: causal GQA flash attention, fp32 WMMA, TDM double-buffered K/V.
// grid = (S/128, B*H); block = 256 (8 waves). Each wave: 16 query rows.
// ---------------------------------------------------------------------------
__global__ __launch_bounds__(256)
void flash_attn_kernel(const float* __restrict__ q_out,
                       const float* __restrict__ k_out,
                       const float* __restrict__ v_out,
                       float* __restrict__ attn)
{
  __shared__ float Klds[2][16][130];   // 16 keys x 128 d, TDM-padded 128->130
  __shared__ float Vlds[2][16][130];
  __shared__ float Pbuf[8][16][17];    // per-wave P transpose buffer

  const int tid  = threadIdx.x;
  const int wave = tid >> 5;
  const int lane = tid & 31;
  const int lo   = lane & 15;
  const int hi   = lane >> 4;

  const int qb  = blockIdx.x * 128;
  const int bh  = blockIdx.y;
  const int b   = bh / H_;
  const int h   = bh % H_;
  const int kvh = h / GROUPS_;

  const float scale = 0.08838834764831845f;   // 1/sqrt(128)
  const float* Qbase = q_out + ((size_t)b * H_   + h)   * S_ * HD_;
  const float* Kbase = k_out + ((size_t)b * KVH_ + kvh) * S_ * HD_;
  const float* Vbase = v_out + ((size_t)b * KVH_ + kvh) * S_ * HD_;
  float*       Obase = attn  + ((size_t)b * H_   + h)   * S_ * HD_;

  const int qrow0 = qb + wave * 16;

  const int nchunks = qb / 16 + 8;   // keys needed: [0, qb+128), 16 per chunk

  // prologue: TDM-load key chunk 0
  if (wave == 0) {
    tdm_load_2d((unsigned)(size_t)&Klds[0][0][0], Kbase,
                128, 16, HD_, S_, HD_, 6, 1);
    tdm_load_2d((unsigned)(size_t)&Vlds[0][0][0], Vbase,
                128, 16, HD_, S_, HD_, 6, 1);
  }

  // Q (16x128) as WMMA A-fragments in registers, pre-scaled.
  float qa[64];
  {
    const float* qrow = Qbase + (size_t)(qrow0 + lo) * HD_;
    #pragma unroll
    for (int kc = 0; kc < 32; ++kc) {
      const int k = kc * 4 + hi * 2;
      qa[2 * kc]     = qrow[k]     * scale;
      qa[2 * kc + 1] = qrow[k + 1] * scale;
    }
  }

  v8f O[8];
  #pragma unroll
  for (int t = 0; t < 8; ++t) O[t] = v8zero();
  float mrow[8], lrow[8];
  #pragma unroll
  for (int r = 0; r < 8; ++r) { mrow[r] = -INFINITY; lrow[r] = 0.f; }

  for (int ch = 0; ch < nchunks; ++ch) {
    const int buf = ch & 1;
    const int k0  = ch * 16;
    if (wave == 0) {
      if (ch + 1 < nchunks) {   // prefetch next 16 keys into other buffer
        const int kn = (ch + 1) * 16;
        tdm_load_2d((unsigned)(size_t)&Klds[buf ^ 1][0][0],
                    Kbase + (size_t)kn * HD_, 128, 16, HD_, S_, HD_, 6, 1);
        tdm_load_2d((unsigned)(size_t)&Vlds[buf ^ 1][0][0],
                    Vbase + (size_t)kn * HD_, 128, 16, HD_, S_, HD_, 6, 1);
        __builtin_amdgcn_s_wait_tensorcnt(2);
      } else {
        __builtin_amdgcn_s_wait_tensorcnt(0);
      }
    }
    __syncthreads();

    if (k0 <= qrow0 + 15) {               // causal: skip fully-masked tiles
      // ---- S = Q * K^T (16x16), 32 fp32 WMMAs over d=128 ----
      v8f sacc = v8zero();
      #pragma unroll
      for (int kc = 0; kc < 32; ++kc) {
        v2f a;  a.x = qa[2 * kc];  a.y = qa[2 * kc + 1];
        const int kd = kc * 4 + hi * 2;
        v2f bf;                            // B[kd][key] = K[key][kd]
        bf.x = Klds[buf][lo][kd];
        bf.y = Klds[buf][lo][kd + 1];
        sacc = __builtin_amdgcn_wmma_f32_16x16x4_f32(
            false, a, false, bf, (short)0, sacc, false, false);
      }
      // ---- mask + online softmax (row r+8*hi lives in one 16-lane half) ----
      #pragma unroll
      for (int r = 0; r < 8; ++r) {
        const int q = qrow0 + r + 8 * hi;
        float sv = ((k0 + lo) <= q) ? sacc[r] : -INFINITY;
        float mx = sv;
        mx = fmaxf(mx, __shfl_xor(mx, 1, 32));
        mx = fmaxf(mx, __shfl_xor(mx, 2, 32));
        mx = fmaxf(mx, __shfl_xor(mx, 4, 32));
        mx = fmaxf(mx, __shfl_xor(mx, 8, 32));
        const float mnew = fmaxf(mrow[r], mx);
        const float corr = __expf(mrow[r] - mnew);
        const float pv   = __expf(sv - mnew);
        float sum = pv;
        sum += __shfl_xor(sum, 1, 32);
        sum += __shfl_xor(sum, 2, 32);
        sum += __shfl_xor(sum, 4, 32);
        sum += __shfl_xor(sum, 8, 32);
        lrow[r] = lrow[r] * corr + sum;
        mrow[r] = mnew;
        #pragma unroll
        for (int t = 0; t < 8; ++t) O[t][r] *= corr;
        Pbuf[wave][r + 8 * hi][lo] = pv;   // C-layout -> LDS
      }
      __builtin_amdgcn_wave_barrier();     // DS ops in-order within wave
      // ---- O += P * V ----
      #pragma unroll
      for (int kc2 = 0; kc2 < 4; ++kc2) {
        v2f a;                             // A-layout read (transposed)
        a.x = Pbuf[wave][lo][kc2 * 4 + hi * 2];
        a.y = Pbuf[wave][lo][kc2 * 4 + hi * 2 + 1];
        const int key = kc2 * 4 + hi * 2;
        #pragma unroll
        for (int t = 0; t < 8; ++t) {
          v2f bf;                          // B[key][d]
          bf.x = Vlds[buf][key][t * 16 + lo];
          bf.y = Vlds[buf][key + 1][t * 16 + lo];
          O[t] = __builtin_amdgcn_wmma_f32_16x16x4_f32(
              false, a, false, bf, (short)0, O[t], false, false);
        }
      }
    }
    __syncthreads();   // reads done before TDM overwrites this buffer
  }

  // ---- finalize: O / l ----
  #pragma unroll
  for (int r = 0; r < 8; ++r) {
    const float inv = 1.f / lrow[r];
    const int row = qrow0 + r + 8 * hi;
    float* orow = Obase + (size_t)row * HD_;
    #pragma unroll
    for (int t = 0; t < 8; ++t) orow[t * 16 + lo] = O[t][r] * inv;
  }
}

// ---------------------------------------------------------------------------
extern "C" void kernel_launch(void* const* d_in, const int* in_sizes, int n_in,
                              void* d_out, int out_size, void* d_ws, size_t ws_size,
                              hipStream_t stream)
{
  (void)in_sizes; (void)n_in; (void)out_size; (void)d_ws; (void)ws_size;

  const float* hidden = (const float*)d_in[0];
  const float* cosT   = (const float*)d_in[1];
  const float* sinT   = (const float*)d_in[2];
  const float* Wq     = (const float*)d_in[3];
  const float* Wk     = (const float*)d_in[4];
  const float* Wv     = (const float*)d_in[5];

  // d_out layout: attn(B,H,S,HD) | q(B,H,S,HD) | k(B,KVH,S,HD) | v(B,KVH,S,HD)
  float* attn  = (float*)d_out;
  float* q_out = attn  + (size_t)B_ * H_   * S_ * HD_;
  float* k_out = q_out + (size_t)B_ * H_   * S_ * HD_;
  float* v_out = k_out + (size_t)B_ * KVH_ * S_ * HD_;

  dim3 g1(B_ * S_ / 128, 24);
  qkv_rope_kernel<<<g1, 256, 0, stream>>>(hidden, cosT, sinT, Wq, Wk, Wv,
                                          q_out, k_out, v_out);
  dim3 g2(S_ / 128, B_ * H_);
  flash_attn_kernel<<<g2, 256, 0, stream>>>(q_out, k_out, v_out, attn);
}